// MultiScaleAttention_22368189677628
// MI455X (gfx1250) — compile-verified
//
#include <hip/hip_runtime.h>
#include <math.h>

// ---------------------------------------------------------------------------
// MultiScaleAttention (MViT block) for MI455X / gfx1250, fp32 via
// V_WMMA_F32_16X16X4_F32 (full precision, matrix pipe).
// B=2 NH=4 T=8 H=14 W=14 DIM=384 HD=96 N=1568
// GEMMs use 32x32 register-blocked tiles: 1 global load per WMMA issue.
// ---------------------------------------------------------------------------

typedef float v2f __attribute__((ext_vector_type(2)));
typedef float v8f __attribute__((ext_vector_type(8)));

#define B_   2
#define NH_  4
#define T_   8
#define H_   14
#define W_   14
#define DIM_ 384
#define HD_  96
#define N_   1568          // T*H*W
#define MR_  (B_ * N_)     // 3136 rows for token-major GEMMs
#define HW_  (H_ * W_)     // 196

static __device__ __forceinline__ v8f wmma_f32(v2f a, v2f b, v8f c) {
  // D(16x16) = A(16x4) * B(4x16) + C, fp32, wave32
  return __builtin_amdgcn_wmma_f32_16x16x4_f32(
      /*neg_a=*/false, a, /*neg_b=*/false, b,
      /*c_mod=*/(short)0, c, /*reuse_a=*/false, /*reuse_b=*/false);
}

// ---------------------------------------------------------------------------
// 1) QKV projection: qkv = x @ qkv_w^T, scattered into [B,NH,N,HD] q/k/v.
//    One wave per 32x32 output tile (2x2 WMMA blocking). K=384.
// ---------------------------------------------------------------------------
__global__ __launch_bounds__(32)
void qkv_gemm_kernel(const float* __restrict__ x, const float* __restrict__ w,
                     float* __restrict__ q, float* __restrict__ k,
                     float* __restrict__ v) {
  const int NT = (3 * DIM_) / 32;              // 36 col tiles
  const int mt = blockIdx.x / NT;
  const int nt = blockIdx.x % NT;
  const int lane = threadIdx.x;
  const int ql = lane & 15;
  const int hi = lane >> 4;                    // 0 or 1

  const float* __restrict__ ap0 = x + (size_t)(mt * 32 + ql) * DIM_ + 2 * hi;
  const float* __restrict__ ap1 = ap0 + (size_t)16 * DIM_;
  const float* __restrict__ bp0 = w + (size_t)(nt * 32 + ql) * DIM_ + 2 * hi;
  const float* __restrict__ bp1 = bp0 + (size_t)16 * DIM_;

  v8f acc00 = {}, acc01 = {}, acc10 = {}, acc11 = {};
  for (int kk = 0; kk < DIM_; kk += 4) {
    v2f a0 = {ap0[kk], ap0[kk + 1]};
    v2f a1 = {ap1[kk], ap1[kk + 1]};
    v2f b0 = {bp0[kk], bp0[kk + 1]};
    v2f b1 = {bp1[kk], bp1[kk + 1]};
    acc00 = wmma_f32(a0, b0, acc00);
    acc01 = wmma_f32(a0, b1, acc01);
    acc10 = wmma_f32(a1, b0, acc10);
    acc11 = wmma_f32(a1, b1, acc11);
  }

  float* __restrict__ outs[3] = {q, k, v};
  v8f accs[2][2] = {{acc00, acc01}, {acc10, acc11}};
  for (int nsub = 0; nsub < 2; ++nsub) {
    const int j     = nt * 32 + nsub * 16 + ql;
    const int which = j / DIM_;                // uniform per subtile (32|384)
    const int head  = (j % DIM_) / HD_;        // uniform per subtile (32|96)
    const int c     = j % HD_;
    float* __restrict__ o = outs[which];
    for (int msub = 0; msub < 2; ++msub) {
      for (int r = 0; r < 8; ++r) {
        const int m   = mt * 32 + msub * 16 + hi * 8 + r;
        const int bq  = m / N_;
        const int tok = m % N_;
        o[(((size_t)bq * NH_ + head) * N_ + tok) * HD_ + c] = accs[msub][nsub][r];
      }
    }
  }
}

// ---------------------------------------------------------------------------
// 2) Depthwise 3x3x3 conv (pad 1) + LayerNorm(HD) per token.
//    Block = 96 threads (one channel each), one block per (b*nh, token).
// ---------------------------------------------------------------------------
__global__ __launch_bounds__(96)
void pool_ln_kernel(const float* __restrict__ in, const float* __restrict__ cw,
                    const float* __restrict__ gamma, const float* __restrict__ beta,
                    float* __restrict__ out) {
  const int blk = blockIdx.x;
  const int tok = blk % N_;
  const int bn  = blk / N_;                       // 0..7
  const int ti  = tok / HW_;
  const int rm  = tok % HW_;
  const int hi  = rm / W_;
  const int wi  = rm % W_;
  const int c   = threadIdx.x;

  const float* __restrict__ wc = cw + c * 27;
  float accv = 0.f;
  for (int dt = -1; dt <= 1; ++dt) {
    const int tt = ti + dt;
    if ((unsigned)tt >= (unsigned)T_) continue;
    for (int dh = -1; dh <= 1; ++dh) {
      const int hh = hi + dh;
      if ((unsigned)hh >= (unsigned)H_) continue;
      for (int dw = -1; dw <= 1; ++dw) {
        const int ww = wi + dw;
        if ((unsigned)ww >= (unsigned)W_) continue;
        const int stok = (tt * H_ + hh) * W_ + ww;
        accv += in[((size_t)bn * N_ + stok) * HD_ + c] *
                wc[(dt + 1) * 9 + (dh + 1) * 3 + (dw + 1)];
      }
    }
  }

  __shared__ float ssum[HD_];
  __shared__ float ssq[HD_];
  ssum[c] = accv;
  ssq[c]  = accv * accv;
  __syncthreads();
  for (int st = 48; st >= 3; st >>= 1) {          // 48,24,12,6,3
    if (c < st) { ssum[c] += ssum[c + st]; ssq[c] += ssq[c + st]; }
    __syncthreads();
  }
  const float tot  = ssum[0] + ssum[1] + ssum[2];
  const float totq = ssq[0] + ssq[1] + ssq[2];
  const float mean = tot * (1.0f / HD_);
  const float var  = totq * (1.0f / HD_) - mean * mean;
  const float rstd = rsqrtf(var + 1e-6f);
  out[((size_t)bn * N_ + tok) * HD_ + c] = (accv - mean) * rstd * gamma[c] + beta[c];
}

// ---------------------------------------------------------------------------
// 3) Decomposed rel-pos dot products:
//    relT[b,n,q,kt] = <q_pooled[b,n,q,:], Rt[ti-kt+T-1,:]>, same for H/W.
// ---------------------------------------------------------------------------
__global__ __launch_bounds__(64)
void relpos_kernel(const float* __restrict__ qp,
                   const float* __restrict__ Rt, const float* __restrict__ Rh,
                   const float* __restrict__ Rw,
                   float* __restrict__ relT, float* __restrict__ relH,
                   float* __restrict__ relW) {
  const int bnq = blockIdx.x;                     // 0 .. 8*1568-1
  const int tok = bnq % N_;
  const int ti  = tok / HW_;
  const int rm  = tok % HW_;
  const int hi  = rm / W_;
  const int wi  = rm % W_;
  const int tid = threadIdx.x;
  const float* __restrict__ qv = qp + (size_t)bnq * HD_;

  if (tid < T_) {
    const int kt = tid;
    const float* rp = Rt + (ti - kt + (T_ - 1)) * HD_;
    float s = 0.f;
    for (int c = 0; c < HD_; ++c) s += qv[c] * rp[c];
    relT[(size_t)bnq * T_ + kt] = s;
  } else if (tid < T_ + H_) {
    const int kh = tid - T_;
    const float* rp = Rh + (hi - kh + (H_ - 1)) * HD_;
    float s = 0.f;
    for (int c = 0; c < HD_; ++c) s += qv[c] * rp[c];
    relH[(size_t)bnq * H_ + kh] = s;
  } else if (tid < T_ + H_ + W_) {
    const int kw = tid - T_ - H_;
    const float* rp = Rw + (wi - kw + (W_ - 1)) * HD_;
    float s = 0.f;
    for (int c = 0; c < HD_; ++c) s += qv[c] * rp[c];
    relW[(size_t)bnq * W_ + kw] = s;
  }
}

// ---------------------------------------------------------------------------
// 4) Fused flash attention + residual. 1 wave per (head, 16-query tile).
//    S^T tiles (k = M dim) so softmax stats reduce with one shfl_xor(16).
//    P tile re-laid through 1KB LDS for the P@V WMMA (out accumulated as out^T).
//    Next K/V tiles prefetched (global_prefetch_b8) under the current WMMAs.
// ---------------------------------------------------------------------------
__global__ __launch_bounds__(32)
void attention_kernel(const float* __restrict__ qp, const float* __restrict__ kp,
                      const float* __restrict__ vp,
                      const float* __restrict__ relT, const float* __restrict__ relH,
                      const float* __restrict__ relW, float* __restrict__ out) {
  const int QT = N_ / 16;                          // 98
  const int bn = blockIdx.x / QT;                  // head index 0..7
  const int qt = blockIdx.x % QT;
  const int lane = threadIdx.x;
  const int ql = lane & 15;
  const int hi = lane >> 4;
  const float scale = 0.1020620726159657f;         // 96^-0.5
  const size_t base = (size_t)bn * N_ * HD_;
  const size_t qrow0 = (size_t)bn * N_ + qt * 16;

  __shared__ float bT[16 * T_];
  __shared__ float bH[16 * H_];
  __shared__ float bW[16 * W_];
  __shared__ float pbuf[16 * 16];

  for (int i = lane; i < 16 * T_; i += 32) {       // 128/32 = 4 iters, no divergence
    bT[i] = relT[(qrow0 + i / T_) * T_ + (i % T_)];
  }
  for (int i = lane; i < 16 * H_; i += 32) {       // 224/32 = 7 iters
    bH[i] = relH[(qrow0 + i / H_) * H_ + (i % H_)];
    bW[i] = relW[(qrow0 + i / W_) * W_ + (i % W_)];
  }
  __syncthreads();

  // Q^T B-fragments (pre-scaled): B[cc, q] = scale * Q[q, cc]
  v2f qb[HD_ / 4];
  {
    const float* __restrict__ qr = qp + base + (size_t)(qt * 16 + ql) * HD_ + 2 * hi;
    for (int cb = 0; cb < HD_ / 4; ++cb) {
      v2f t = {qr[cb * 4] * scale, qr[cb * 4 + 1] * scale};
      qb[cb] = t;
    }
  }

  v8f acc[6] = {};                                  // out^T: 6 channel tiles
  float mrun = -INFINITY, lrun = 0.f;

  for (int kt_ = 0; kt_ < QT; ++kt_) {
    // prefetch next K/V tile rows into cache while this tile computes
    if (kt_ + 1 < QT) {
      const float* nk = kp + base + (size_t)((kt_ + 1) * 16 + ql) * HD_ + 2 * hi;
      const float* nv = vp + base + (size_t)((kt_ + 1) * 16 + ql) * HD_;
      __builtin_prefetch(nk, 0, 1);
      __builtin_prefetch(nv, 0, 1);
    }

    // --- S^T tile = K_tile(16x96) @ Q^T(96x16), fp32 WMMA, K-chunks of 4
    v8f st = {};
    const float* __restrict__ kr = kp + base + (size_t)(kt_ * 16 + ql) * HD_ + 2 * hi;
    for (int cb = 0; cb < HD_ / 4; ++cb) {
      v2f a = {kr[cb * 4], kr[cb * 4 + 1]};
      st = wmma_f32(a, qb[cb], st);
    }

    // --- add decomposed rel-pos bias; track tile max (per query column)
    float sv[8];
    float tmax = -INFINITY;
    for (int r = 0; r < 8; ++r) {
      const int kg  = kt_ * 16 + hi * 8 + r;        // key index
      const int kt2 = kg / HW_;
      const int krm = kg % HW_;
      const float val = st[r] + bT[ql * T_ + kt2] + bH[ql * H_ + krm / W_] +
                        bW[ql * W_ + krm % W_];
      sv[r] = val;
      tmax = fmaxf(tmax, val);
    }
    tmax = fmaxf(tmax, __shfl_xor(tmax, 16, 32));   // join lane halves (same q)

    // --- online softmax update
    const float mnew  = fmaxf(mrun, tmax);
    const float alpha = __expf(mrun - mnew);
    float psum = 0.f;
    for (int r = 0; r < 8; ++r) {
      sv[r] = __expf(sv[r] - mnew);
      psum += sv[r];
    }
    psum += __shfl_xor(psum, 16, 32);
    lrun = lrun * alpha + psum;
    mrun = mnew;
    for (int ct = 0; ct < 6; ++ct)
      for (int r = 0; r < 8; ++r) acc[ct][r] *= alpha;

    // --- P^T tile -> LDS for B-operand re-layout
    __syncthreads();
    for (int r = 0; r < 8; ++r) pbuf[(hi * 8 + r) * 16 + ql] = sv[r];
    __syncthreads();

    // --- out^T += V^T_tile(96x16) @ P^T(16x16): 6 c-tiles x 4 K-chunks
    for (int ct = 0; ct < 6; ++ct) {
      for (int kc = 0; kc < 4; ++kc) {
        const int k0 = kc * 4 + 2 * hi;             // local key index
        v2f a = {vp[base + (size_t)(kt_ * 16 + k0) * HD_ + ct * 16 + ql],
                 vp[base + (size_t)(kt_ * 16 + k0 + 1) * HD_ + ct * 16 + ql]};
        v2f b = {pbuf[k0 * 16 + ql], pbuf[(k0 + 1) * 16 + ql]};
        acc[ct] = wmma_f32(a, b, acc[ct]);
      }
    }
  }

  // --- epilogue: normalize, residual (+ pooled q), store [B,NH,N,HD]
  const float rl = 1.f / lrun;
  const int qg = qt * 16 + ql;
  for (int ct = 0; ct < 6; ++ct) {
    for (int r = 0; r < 8; ++r) {
      const int c = ct * 16 + hi * 8 + r;
      const size_t idx = base + (size_t)qg * HD_ + c;
      out[idx] = acc[ct][r] * rl + qp[idx];
    }
  }
}

// ---------------------------------------------------------------------------
// 5) Output projection: out = concat_heads(attn_out) @ proj_w^T + proj_b.
//    32x32 register-blocked tiles, head-gather on A operand.
// ---------------------------------------------------------------------------
__global__ __launch_bounds__(32)
void proj_gemm_kernel(const float* __restrict__ ain, const float* __restrict__ w,
                      const float* __restrict__ bias, float* __restrict__ out) {
  const int NT = DIM_ / 32;                        // 12 col tiles
  const int mt = blockIdx.x / NT;
  const int nt = blockIdx.x % NT;
  const int lane = threadIdx.x;
  const int ql = lane & 15;
  const int hi = lane >> 4;

  const int m0  = mt * 32 + ql;                    // A rows for the two subtiles
  const int m1  = m0 + 16;
  const int bq0 = m0 / N_, tk0 = m0 % N_;
  const int bq1 = m1 / N_, tk1 = m1 % N_;
  const float* __restrict__ bp0 = w + (size_t)(nt * 32 + ql) * DIM_ + 2 * hi;
  const float* __restrict__ bp1 = bp0 + (size_t)16 * DIM_;

  v8f acc00 = {}, acc01 = {}, acc10 = {}, acc11 = {};
  for (int kk = 0; kk < DIM_; kk += 4) {
    const int k0 = kk + 2 * hi;                    // even -> never crosses head
    const int hd = k0 / HD_;
    const int c0 = k0 % HD_;
    const size_t ar0 = (((size_t)bq0 * NH_ + hd) * N_ + tk0) * HD_ + c0;
    const size_t ar1 = (((size_t)bq1 * NH_ + hd) * N_ + tk1) * HD_ + c0;
    v2f a0 = {ain[ar0], ain[ar0 + 1]};
    v2f a1 = {ain[ar1], ain[ar1 + 1]};
    v2f b0 = {bp0[kk], bp0[kk + 1]};
    v2f b1 = {bp1[kk], bp1[kk + 1]};
    acc00 = wmma_f32(a0, b0, acc00);
    acc01 = wmma_f32(a0, b1, acc01);
    acc10 = wmma_f32(a1, b0, acc10);
    acc11 = wmma_f32(a1, b1, acc11);
  }

  v8f accs[2][2] = {{acc00, acc01}, {acc10, acc11}};
  for (int nsub = 0; nsub < 2; ++nsub) {
    const int j  = nt * 32 + nsub * 16 + ql;
    const float bj = bias[j];
    for (int msub = 0; msub < 2; ++msub) {
      for (int r = 0; r < 8; ++r) {
        const int mm = mt * 32 + msub * 16 + hi * 8 + r;
        out[(size_t)mm * DIM_ + j] = accs[msub][nsub][r] + bj;
      }
    }
  }
}

// ---------------------------------------------------------------------------
extern "C" void kernel_launch(void* const* d_in, const int* in_sizes, int n_in,
                              void* d_out, int out_size, void* d_ws, size_t ws_size,
                              hipStream_t stream) {
  const float* x       = (const float*)d_in[0];
  const float* qkv_w   = (const float*)d_in[1];
  const float* proj_w  = (const float*)d_in[2];
  const float* proj_b  = (const float*)d_in[3];
  const float* pool_qw = (const float*)d_in[4];
  const float* pool_kw = (const float*)d_in[5];
  const float* pool_vw = (const float*)d_in[6];
  const float* nq_w    = (const float*)d_in[7];
  const float* nq_b    = (const float*)d_in[8];
  const float* nk_w    = (const float*)d_in[9];
  const float* nk_b    = (const float*)d_in[10];
  const float* nv_w    = (const float*)d_in[11];
  const float* nv_b    = (const float*)d_in[12];
  const float* rel_h   = (const float*)d_in[13];
  const float* rel_w   = (const float*)d_in[14];
  const float* rel_t   = (const float*)d_in[15];

  float* ws = (float*)d_ws;
  const size_t PH = (size_t)B_ * NH_ * N_ * HD_;   // 1,204,224 floats
  float* qS   = ws;
  float* kS   = ws + PH;
  float* vS   = ws + 2 * PH;
  float* qpS  = ws + 3 * PH;
  float* kpS  = ws + 4 * PH;
  float* vpS  = ws + 5 * PH;
  float* aoS  = ws + 6 * PH;
  float* rTS  = ws + 7 * PH;                       // [B,NH,N,8]
  float* rHS  = rTS + (size_t)B_ * NH_ * N_ * T_;  // [B,NH,N,14]
  float* rWS  = rHS + (size_t)B_ * NH_ * N_ * H_;  // [B,NH,N,14]

  // 1) fused QKV projection (WMMA f32, 32x32 tiles)
  qkv_gemm_kernel<<<(MR_ / 32) * ((3 * DIM_) / 32), 32, 0, stream>>>(
      x, qkv_w, qS, kS, vS);

  // 2) depthwise conv3d + LayerNorm for q/k/v
  const int nPool = B_ * NH_ * N_;
  pool_ln_kernel<<<nPool, HD_, 0, stream>>>(qS, pool_qw, nq_w, nq_b, qpS);
  pool_ln_kernel<<<nPool, HD_, 0, stream>>>(kS, pool_kw, nk_w, nk_b, kpS);
  pool_ln_kernel<<<nPool, HD_, 0, stream>>>(vS, pool_vw, nv_w, nv_b, vpS);

  // 3) decomposed rel-pos terms from pooled q
  relpos_kernel<<<B_ * NH_ * N_, 64, 0, stream>>>(qpS, rel_t, rel_h, rel_w,
                                                  rTS, rHS, rWS);

  // 4) fused flash attention + residual (WMMA f32)
  attention_kernel<<<B_ * NH_ * (N_ / 16), 32, 0, stream>>>(
      qpS, kpS, vpS, rTS, rHS, rWS, aoS);

  // 5) output projection (WMMA f32, 32x32 tiles)
  proj_gemm_kernel<<<(MR_ / 32) * (DIM_ / 32), 32, 0, stream>>>(
      aoS, proj_w, proj_b, (float*)d_out);
}